// EqAMPBC_43508018709155
// MI455X (gfx1250) — compile-verified
//
#include <hip/hip_runtime.h>

typedef float v2f __attribute__((ext_vector_type(2)));
typedef float v8f __attribute__((ext_vector_type(8)));

#define HDIM 260
#define MDIM 41
#define PHALF 20
#define RSTRIDE 170      // dwords per b-region in LDS (odd*2 stride -> conflict-free b64)
#define REG_PER_BLK 128  // 8 waves * 16 batch rows
#define THREADS 256

// ---- compile-time fwm_index tables (must match reference loop order) ----
constexpr int count_pairs() {
  int k = 0;
  for (int m = -PHALF; m <= PHALF; ++m)
    for (int n = -PHALF; n <= PHALF; ++n) {
      if (m == 0 || n == 0) continue;
      int mn = m * n; mn = mn < 0 ? -mn : mn;
      int s = m + n;  s  = s  < 0 ? -s  : s;
      if (mn <= PHALF && s <= PHALF) ++k;
    }
  return k;
}
static_assert(count_pairs() == HDIM, "HDIM mismatch with fwm_index");

struct alignas(16) OffTab { unsigned v[HDIM]; };

constexpr OffTab build_tab() {
  OffTab t{};
  int k = 0;
  for (int m = -PHALF; m <= PHALF; ++m)
    for (int n = -PHALF; n <= PHALF; ++n) {
      if (m == 0 || n == 0) continue;
      int mn = m * n; mn = mn < 0 ? -mn : mn;
      int s = m + n;  s  = s  < 0 ? -s  : s;
      if (mn <= PHALF && s <= PHALF) {
        unsigned a = (unsigned)(PHALF + n);       // p + N_IDX
        unsigned b = (unsigned)(PHALF + m + n);   // p + M_IDX + N_IDX
        unsigned c = (unsigned)(PHALF + m);       // p + M_IDX
        t.v[k++] = a | (b << 6) | (c << 12);
      }
    }
  return t;
}
__constant__ OffTab g_tab = build_tab();

// LDS region layout per batch element: dword = pos*4 + mode*2 + {0:re,1:im}
__global__ __launch_bounds__(THREADS)
void eqampbc_kernel(const float* __restrict__ xre, const float* __restrict__ xim,
                    const float* __restrict__ task, const float* __restrict__ c00p,
                    const float* __restrict__ fwm_wr, const float* __restrict__ fwm_wi,
                    const float* __restrict__ w1p, const float* __restrict__ w2p,
                    float* __restrict__ out)
{
  extern __shared__ float smem[];
  float* ldsX = smem;                               // 128 * 170 dwords
  float* ldsD = smem + REG_PER_BLK * RSTRIDE;       // 8 waves * 64 dwords

  const int tid  = threadIdx.x;
  const int lane = tid & 31;
  const int w    = tid >> 5;          // wave id in block
  const int g    = lane >> 4;         // K-half / h-slot selector
  const int r    = lane & 15;         // batch row within wave tile
  const long long b0 = (long long)blockIdx.x * REG_PER_BLK;

  // ---- stage raw x into LDS, interleaving re/im per region ----
  for (int u = tid; u < REG_PER_BLK * MDIM; u += THREADS) {
    const int region = u / MDIM;
    const int q = u - region * MDIM;                 // position m
    const long long src = (b0 + region) * (MDIM * 2) + q * 2;
    const float2 re2 = *(const float2*)(xre + src);  // modes 0,1 real
    const float2 im2 = *(const float2*)(xim + src);  // modes 0,1 imag
    float* d = ldsX + region * RSTRIDE + q * 4;
    *(float2*)(d)     = make_float2(re2.x, im2.x);   // mode0 (re,im)
    *(float2*)(d + 2) = make_float2(re2.y, im2.y);   // mode1 (re,im)
  }
  __syncthreads();

  const float* myX = ldsX + (w * 16 + r) * RSTRIDE;

  // one-hot selectors to place W into B-tile lanes (cols 0..3)
  const float a0 = (r == 0) ? 1.f : 0.f;
  const float a1 = (r == 1) ? 1.f : 0.f;
  const float a2 = (r == 2) ? 1.f : 0.f;
  const float a3 = (r == 3) ? 1.f : 0.f;

  v8f acc = {0.f, 0.f, 0.f, 0.f, 0.f, 0.f, 0.f, 0.f};

  for (int it = 0; it < HDIM / 2; ++it) {
    // uniform scalar loads for both h of this step, then per-half select
    const uint2 pk2 = *(const uint2*)(g_tab.v + 2 * it);
    const unsigned pk = g ? pk2.y : pk2.x;
    const int offA = (int)(pk & 63u);
    const int offB = (int)((pk >> 6) & 63u);
    const int offC = (int)((pk >> 12) & 63u);

    const float2 wrA = *(const float2*)(fwm_wr + 2 * it);          // mode0 Wre h0,h1
    const float2 wrB = *(const float2*)(fwm_wr + HDIM + 2 * it);   // mode1 Wre
    const float2 wiA = *(const float2*)(fwm_wi + 2 * it);          // mode0 Wim
    const float2 wiB = *(const float2*)(fwm_wi + HDIM + 2 * it);   // mode1 Wim
    const float wr0 = g ? wrA.y : wrA.x;
    const float wr1 = g ? wrB.y : wrB.x;
    const float wi0 = g ? wiA.y : wiA.x;
    const float wi1 = g ? wiB.y : wiB.x;

    // gathers (raw, unscaled x)
    const float2 en0  = *(const float2*)(myX + offA * 4);
    const float2 en1  = *(const float2*)(myX + offA * 4 + 2);
    const float2 emn0 = *(const float2*)(myX + offB * 4);
    const float2 emn1 = *(const float2*)(myX + offB * 4 + 2);
    const float2 em0  = *(const float2*)(myX + offC * 4);
    const float2 em1  = *(const float2*)(myX + offC * 4 + 2);

    // S = sum over modes of En * conj(Emn)  (same for both modes after roll-add)
    const float sre = en0.x*emn0.x + en0.y*emn0.y + en1.x*emn1.x + en1.y*emn1.y;
    const float sim = en0.y*emn0.x - en0.x*emn0.y + en1.y*emn1.x - en1.x*emn1.y;

    // F[mode] = S * Em[mode]
    v2f A0, A1, B0, B1;
    A0[0] = sre*em0.x - sim*em0.y;  A0[1] = sre*em0.y + sim*em0.x;
    A1[0] = sre*em1.x - sim*em1.y;  A1[1] = sre*em1.y + sim*em1.x;

    // B encodes complex multiply by W[n,h]; cols: 0=re0 1=im0 2=re1 3=im1
    B0[0] = a0*wr0 + a1*wi0;  B0[1] = a1*wr0 - a0*wi0;
    B1[0] = a2*wr1 + a3*wi1;  B1[1] = a3*wr1 - a2*wi1;

    // E[b,c] += F * W  on the matrix pipe (K = {re,im} x {h0,h1})
    acc = __builtin_amdgcn_wmma_f32_16x16x4_f32(false, A0, false, B0, (short)0, acc, false, false);
    acc = __builtin_amdgcn_wmma_f32_16x16x4_f32(false, A1, false, B1, (short)0, acc, false, false);
  }

  // spill D (rows = local b, cols 0..3) through per-wave LDS scratch
  {
    const int col = lane & 15;
    const int rowBase = (lane < 16) ? 0 : 8;
    if (col < 4) {
#pragma unroll
      for (int j = 0; j < 8; ++j)
        ldsD[w * 64 + (rowBase + j) * 4 + col] = acc[j];
    }
  }
  __syncthreads();

  // ---- tail: lane = (mode n = g, local b = r) ----
  const int n = g;
  const long long b = b0 + w * 16 + r;
  const float efr = ldsD[w * 64 + r * 4 + 2 * n];
  const float efi = ldsD[w * 64 + r * 4 + 2 * n + 1];

  const float ti = task[b * 4];
  const float P  = __expf(0.2302585092994046f * ti) * 0.5f;  // 10^(ti/10)/2
  const float C00 = c00p[0];

  float accphi = 0.f, cr = 0.f, ci = 0.f;
  float pp = 0.f, xnr = 0.f, xni = 0.f, xor_ = 0.f, xoi = 0.f;
#pragma unroll 1
  for (int m = 0; m < MDIM; ++m) {
    const float2 xn_ = *(const float2*)(myX + m * 4 + n * 2);
    const float2 xo_ = *(const float2*)(myX + m * 4 + (1 - n) * 2);
    const float pn = xn_.x*xn_.x + xn_.y*xn_.y;
    const float po = xo_.x*xo_.x + xo_.y*xo_.y;
    float w1m = w1p[m], w2m = w2p[m];
    if (m == PHALF) {
      w1m = 0.f; w2m = 0.f;
      pp = pn + po;
      xnr = xn_.x; xni = xn_.y; xor_ = xo_.x; xoi = xo_.y;
    }
    accphi += w1m * (2.f * pn + po);
    cr += w2m * (xn_.x*xo_.x + xn_.y*xo_.y);   // x_n * conj(x_other)
    ci += w2m * (xn_.y*xo_.x - xn_.x*xo_.y);
  }

  const float phi = P * (C00 * pp + 2.f * accphi);
  float s, c;
  __sincosf(phi, &s, &c);

  // z = x[p,1-n] * conv ; ixixpm = i*z  -> (-z_im, z_re); all carry factor P
  const float zr = xor_*cr - xoi*ci;
  const float zi = xor_*ci + xoi*cr;
  const float Ere = P * efr - P * zi + (xnr * c - xni * s);
  const float Eim = P * efi + P * zr + (xnr * s + xni * c);

  *(float2*)(out + b * 4 + n * 2) = make_float2(Ere, Eim);
}

extern "C" void kernel_launch(void* const* d_in, const int* in_sizes, int n_in,
                              void* d_out, int out_size, void* d_ws, size_t ws_size,
                              hipStream_t stream) {
  const float* xre  = (const float*)d_in[0];
  const float* xim  = (const float*)d_in[1];
  const float* task = (const float*)d_in[2];
  const float* c00  = (const float*)d_in[3];
  const float* wr   = (const float*)d_in[4];
  const float* wi   = (const float*)d_in[5];
  const float* w1   = (const float*)d_in[6];
  const float* w2   = (const float*)d_in[7];
  float* outp = (float*)d_out;

  const int B = in_sizes[0] / (MDIM * 2);   // 131072; assumed multiple of 128
  const int grid = B / REG_PER_BLK;
  const size_t shmem = (size_t)(REG_PER_BLK * RSTRIDE + 8 * 64) * sizeof(float);
  eqampbc_kernel<<<grid, THREADS, shmem, stream>>>(xre, xim, task, c00, wr, wi, w1, w2, outp);
}